// Convolution_61804579389712
// MI455X (gfx1250) — compile-verified
//
#include <hip/hip_runtime.h>
#include <math.h>

#define NEDGES 800000
#define NODEF  30
#define RCUT   5.0f
#define WAVES  4
#define EPB    (WAVES*16)   // edges per block

typedef __attribute__((ext_vector_type(2))) float v2f;
typedef __attribute__((ext_vector_type(8))) float v8f;

__global__ __launch_bounds__(128, 1)
void edge_kernel(const float* __restrict__ nodes, const float* __restrict__ pos,
                 const float* __restrict__ Wr,    const float* __restrict__ br,
                 const int*   __restrict__ src,   const int*   __restrict__ dst,
                 float* __restrict__ out)
{
    __shared__ float lds_w[WAVES][16][224];
    __shared__ float lds_y[WAVES][16][8];
    __shared__ float lds_env[WAVES][16];
    __shared__ int   lds_src[WAVES][16];
    __shared__ int   lds_dst[WAVES][16];

    const int tid  = threadIdx.x;
    const int wv   = tid >> 5;
    const int lane = tid & 31;
    const int half = lane >> 4;
    const int hl   = lane & 15;
    const int tileBase = blockIdx.x * EPB + wv * 16;

    // ---------------- Phase 1: per-edge geometry, bessel, SH (lanes l and l+16 redundantly do edge l)
    int e  = tileBase + hl;
    int ec = (e < NEDGES) ? e : (NEDGES - 1);
    int si = src[ec], di = dst[ec];
    float rx = pos[si*3+0] - pos[di*3+0];
    float ry = pos[si*3+1] - pos[di*3+1];
    float rz = pos[si*3+2] - pos[di*3+2];
    float dist = sqrtf(rx*rx + ry*ry + rz*rz + 1e-24f);
    float inv  = 1.0f / dist;
    float gate = (dist < RCUT) ? 1.0f : 0.0f;

    float bess[12];
    {
        float pref = 0.6324555320f * gate * inv;          // sqrt(2/RCUT)
        float t  = 3.14159265358979f * dist * (1.0f / RCUT);
        float s1 = sinf(t), c1 = cosf(t);
        float twoc = 2.0f * c1, skm = 0.0f, sk = s1;
        bess[0] = pref * sk;
        #pragma unroll
        for (int k = 1; k < 10; k++) { float sn = twoc*sk - skm; skm = sk; sk = sn; bess[k] = pref*sk; }
        bess[10] = 0.0f; bess[11] = 0.0f;
    }
    float uu = dist * (1.0f / RCUT);
    float u2 = uu*uu, u5 = u2*u2*uu, u6 = u5*uu;
    float env = (1.0f - 6.0f*u5 + 5.0f*u6) * gate;
    float xc = rx*inv, yc = ry*inv, zc = rz*inv;

    if (half == 0) {
        lds_src[wv][hl] = si;  lds_dst[wv][hl] = di;  lds_env[wv][hl] = env;
        lds_y[wv][hl][0] = 1.73205081f * xc;
        lds_y[wv][hl][1] = 1.73205081f * yc;
        lds_y[wv][hl][2] = 1.73205081f * zc;
        lds_y[wv][hl][3] = 3.87298335f * xc * zc;
        lds_y[wv][hl][4] = 3.87298335f * xc * yc;
        lds_y[wv][hl][5] = 2.23606798f * (yc*yc - 0.5f*(xc*xc + zc*zc));
        lds_y[wv][hl][6] = 3.87298335f * yc * zc;
        lds_y[wv][hl][7] = 0.5f * 3.87298335f * (zc*zc - xc*xc);
    }
    __syncthreads();

    float env8[8];
    #pragma unroll
    for (int r = 0; r < 8; r++) env8[r] = lds_env[wv][half*8 + r];

    // ---------------- Phase 2: radial MLP (16 edges x 216 cols) via f32 WMMA, K=10 padded to 12.
    // All loads are branchless clamped-address loads:
    //  - padded cols (>=216) only pollute their own column -> lands in lds_w padding (never read)
    //  - padded rows K=10,11 multiply A==0 (bess[10..11]=0), any finite B value is harmless
    #pragma unroll 2
    for (int cn = 0; cn < 14; cn++) {
        int col  = cn*16 + hl;
        int colc = (col < 216) ? col : 215;          // clamp, no exec masking
        v8f acc = {};
        #pragma unroll
        for (int ck = 0; ck < 3; ck++) {
            int k0  = ck*4 + half*2;
            int kr0 = (k0     < 10) ? k0     : 9;    // clamp row, contribution is x0 anyway
            int kr1 = (k0 + 1 < 10) ? k0 + 1 : 9;
            v2f a, b;
            a.x = half ? bess[ck*4+2] : bess[ck*4+0];
            a.y = half ? bess[ck*4+3] : bess[ck*4+1];
            b.x = Wr[kr0*216 + colc];
            b.y = Wr[kr1*216 + colc];
            acc = __builtin_amdgcn_wmma_f32_16x16x4_f32(false, a, false, b, (short)0, acc, false, false);
        }
        float bias = br[colc];
        #pragma unroll
        for (int r = 0; r < 8; r++) {
            float h   = acc[r] + bias;
            // silu via fast reciprocal (v_rcp_f32, ~1 ulp) instead of IEEE divide chain
            float act = h * __builtin_amdgcn_rcpf(1.0f + __expf(-h));
            lds_w[wv][half*8 + r][col] = act * env8[r];
        }
    }
    __syncthreads();

    // ---------------- Phase 3: tensor-product paths, 2 lanes per edge
    const int pe   = lane >> 1;
    const int role = lane & 1;
    bool valid = (tileBase + pe) < NEDGES;
    int  s3 = lds_src[wv][pe];
    int  d3 = lds_dst[wv][pe];
    const float* W = &lds_w[wv][pe][0];
    const float* Y = &lds_y[wv][pe][0];
    float y1v[3] = {Y[0], Y[1], Y[2]};
    float y2v[5] = {Y[3], Y[4], Y[5], Y[6], Y[7]};

    // vectorized gather of the 30-float node row (8-byte aligned)
    float xf[30];
    {
        const float2* X2 = reinterpret_cast<const float2*>(nodes + (size_t)s3 * NODEF);
        #pragma unroll
        for (int j = 0; j < 15; j++) { float2 t = X2[j]; xf[2*j] = t.x; xf[2*j+1] = t.y; }
    }
    float x0[8], x1[4][3], x2[2][5];
    #pragma unroll
    for (int i = 0; i < 8; i++) x0[i] = xf[i];
    #pragma unroll
    for (int u = 0; u < 4; u++)
        #pragma unroll
        for (int i = 0; i < 3; i++) x1[u][i] = xf[8 + u*3 + i];
    #pragma unroll
    for (int u = 0; u < 2; u++)
        #pragma unroll
        for (int k = 0; k < 5; k++) x2[u][k] = xf[20 + u*5 + k];
    float* O = out + (size_t)d3 * 30;

    const float PW0 = 0.2672612419f, PW1 = 0.4082482905f, PW2 = 0.5590169944f;
    const float IS3 = 0.5773502692f, IS5 = 0.4472135955f;
    const float CA = 0.3162277660f, CB = 0.3651483717f, CC = -0.1825741858f;   // w3j(1,1,2) family
    const float CD = 0.2390457220f, CE = 0.1195228610f, CF = 0.2070196678f;    // w3j(2,2,2)
    const float SC = 0.2f;                                                     // 1/sqrt(25)

    if (role == 0) {
        // -------- out0 (paths 1,5,10)
        float z5[4], z10[2];
        #pragma unroll
        for (int u = 0; u < 4; u++)
            z5[u] = IS3*(x1[u][0]*y1v[0] + x1[u][1]*y1v[1] + x1[u][2]*y1v[2]);
        #pragma unroll
        for (int u = 0; u < 2; u++) {
            float a = 0.0f;
            #pragma unroll
            for (int k = 0; k < 5; k++) a += x2[u][k]*y2v[k];
            z10[u] = IS5*a;
        }
        #pragma unroll
        for (int w = 0; w < 8; w++) {
            float acc = 0.0f;
            #pragma unroll
            for (int u = 0; u < 8; u++) acc += x0[u]*W[u*8 + w];
            #pragma unroll
            for (int u = 0; u < 4; u++) acc += z5[u]*W[128 + u*8 + w];
            #pragma unroll
            for (int u = 0; u < 2; u++) acc += z10[u]*W[196 + u*8 + w];
            if (valid) atomicAdd(&O[w], SC*PW0*acc);
        }
        // -------- out2 (paths 3,6,8,11)
        float z6[4][5];
        #pragma unroll
        for (int u = 0; u < 4; u++) {
            const float* v = x1[u];
            z6[u][0] = CA*(v[0]*y1v[2] + v[2]*y1v[0]);
            z6[u][1] = CA*(v[0]*y1v[1] + v[1]*y1v[0]);
            z6[u][2] = CC*v[0]*y1v[0] + CB*v[1]*y1v[1] + CC*v[2]*y1v[2];
            z6[u][3] = CA*(v[1]*y1v[2] + v[2]*y1v[1]);
            z6[u][4] = CA*(v[2]*y1v[2] - v[0]*y1v[0]);
        }
        float z11[2][5];
        #pragma unroll
        for (int u = 0; u < 2; u++) {
            const float* Xu = x2[u];
            z11[u][0] =  CD*(Xu[0]*y2v[2] + Xu[2]*y2v[0]) - CF*(Xu[1]*y2v[3] + Xu[3]*y2v[1]);
            z11[u][1] = -CE*(Xu[1]*y2v[2] + Xu[2]*y2v[1]) + CF*(Xu[1]*y2v[4] + Xu[4]*y2v[1])
                        -CF*(Xu[0]*y2v[3] + Xu[3]*y2v[0]);
            z11[u][2] =  CD*Xu[0]*y2v[0] - CE*Xu[1]*y2v[1] - CD*Xu[2]*y2v[2] - CE*Xu[3]*y2v[3];
            z11[u][3] = -CE*(Xu[3]*y2v[2] + Xu[2]*y2v[3]) - CF*(Xu[3]*y2v[4] + Xu[4]*y2v[3])
                        -CF*(Xu[0]*y2v[1] + Xu[1]*y2v[0]);
            z11[u][4] =  CD*(Xu[2]*y2v[4] + Xu[4]*y2v[2]) + CF*Xu[1]*y2v[1] - CF*Xu[3]*y2v[3];
        }
        float s3w[2];
        #pragma unroll
        for (int w = 0; w < 2; w++) {
            float a = 0.0f;
            #pragma unroll
            for (int u = 0; u < 8; u++) a += x0[u]*W[96 + u*2 + w];
            s3w[w] = a;
        }
        #pragma unroll
        for (int w = 0; w < 2; w++)
            #pragma unroll
            for (int k = 0; k < 5; k++) {
                float acc = IS5*s3w[w]*y2v[k];
                #pragma unroll
                for (int u = 0; u < 4; u++) acc += z6[u][k]*W[160 + u*2 + w];
                #pragma unroll
                for (int u = 0; u < 2; u++) acc += IS5*x2[u][k]*W[184 + u*2 + w];
                #pragma unroll
                for (int u = 0; u < 2; u++) acc += z11[u][k]*W[212 + u*2 + w];
                if (valid) atomicAdd(&O[20 + w*5 + k], SC*PW2*acc);
            }
    } else {
        // -------- out1 (paths 2,4,7,9)
        float s2w[4];
        #pragma unroll
        for (int w = 0; w < 4; w++) {
            float a = 0.0f;
            #pragma unroll
            for (int u = 0; u < 8; u++) a += x0[u]*W[64 + u*4 + w];
            s2w[w] = a;
        }
        float A00 = CC*y2v[2] - CA*y2v[4], A11 = CB*y2v[2], A22 = CC*y2v[2] + CA*y2v[4];
        float A01 = CA*y2v[1], A02 = CA*y2v[0], A12 = CA*y2v[3];
        float zA[4][3];
        #pragma unroll
        for (int u = 0; u < 4; u++) {
            const float* v = x1[u];
            zA[u][0] = v[0]*A00 + v[1]*A01 + v[2]*A02;
            zA[u][1] = v[0]*A01 + v[1]*A11 + v[2]*A12;
            zA[u][2] = v[0]*A02 + v[1]*A12 + v[2]*A22;
        }
        float zB[2][3];
        #pragma unroll
        for (int u = 0; u < 2; u++) {
            const float* Xu = x2[u];
            float B00 = CC*Xu[2] - CA*Xu[4], B11 = CB*Xu[2], B22 = CC*Xu[2] + CA*Xu[4];
            float B01 = CA*Xu[1], B02 = CA*Xu[0], B12 = CA*Xu[3];
            zB[u][0] = y1v[0]*B00 + y1v[1]*B01 + y1v[2]*B02;
            zB[u][1] = y1v[0]*B01 + y1v[1]*B11 + y1v[2]*B12;
            zB[u][2] = y1v[0]*B02 + y1v[1]*B12 + y1v[2]*B22;
        }
        #pragma unroll
        for (int w = 0; w < 4; w++)
            #pragma unroll
            for (int k = 0; k < 3; k++) {
                float acc = IS3*s2w[w]*y1v[k];
                #pragma unroll
                for (int u = 0; u < 4; u++) acc += IS3*x1[u][k]*W[112 + u*4 + w];
                #pragma unroll
                for (int u = 0; u < 4; u++) acc += zA[u][k]*W[168 + u*4 + w];
                #pragma unroll
                for (int u = 0; u < 2; u++) acc += zB[u][k]*W[188 + u*4 + w];
                if (valid) atomicAdd(&O[8 + w*3 + k], SC*PW1*acc);
            }
    }
}

extern "C" void kernel_launch(void* const* d_in, const int* in_sizes, int n_in,
                              void* d_out, int out_size, void* d_ws, size_t ws_size,
                              hipStream_t stream) {
    const float* nodes = (const float*)d_in[0];
    const float* pos   = (const float*)d_in[1];
    const float* Wr    = (const float*)d_in[2];
    const float* br    = (const float*)d_in[3];
    const int*   src   = (const int*)d_in[4];
    const int*   dst   = (const int*)d_in[5];
    float* out = (float*)d_out;

    hipMemsetAsync(out, 0, (size_t)out_size * sizeof(float), stream);
    int nb = (NEDGES + EPB - 1) / EPB;   // 12500
    edge_kernel<<<nb, 128, 0, stream>>>(nodes, pos, Wr, br, src, dst, out);
}